// DotProductAttention_20675972563110
// MI455X (gfx1250) — compile-verified
//
#include <hip/hip_runtime.h>
#include <math.h>
#include <stdint.h>

typedef float v2f __attribute__((ext_vector_type(2)));
typedef float v8f __attribute__((ext_vector_type(8)));

#define BATCH 4
#define SEQ   4096
#define DM    64
#define NKT   (SEQ / 16)      // 256 K-tiles per batch
#define BAND_LO (-64)         // j - i >= -ATTN_WIDTH/2
#define BAND_HI 63            // j - i <= ATTN_WIDTH/2 - 1
#define KPITCH 68             // LDS row pitch (floats): bank-conflict-free fragment reads
#define KBUFSZ (16 * KPITCH)  // floats per staged K tile

// D = A(16x4 f32) * B(4x16 f32) + C(16x16 f32), full fp32 precision.
// A layout: lane m=lane&15 holds A[m][kh*2], A[m][kh*2+1] (kh = lane>>4).
// B layout: lane n=lane&15 holds B[kh*2][n], B[kh*2+1][n].
// C/D layout: lane n holds D[r + 8*kh][n] in component r (r=0..7).
__device__ __forceinline__ v8f wmma4(v2f a, v2f b, v8f c) {
  return __builtin_amdgcn_wmma_f32_16x16x4_f32(
      /*neg_a=*/false, a, /*neg_b=*/false, b,
      /*c_mod=*/(short)0, c, /*reuse_a=*/false, /*reuse_b=*/false);
}

// Stage one 16x64 f32 K-tile into LDS (row pitch KPITCH) with 8 coalesced
// 512-byte async-to-LDS transfers.  Tracked by ASYNCcnt.
__device__ __forceinline__ void async_stage_tile(const float* __restrict__ gsrc,
                                                 uint32_t lds_base, int lane) {
#pragma unroll
  for (int i = 0; i < 8; ++i) {
    const int f   = i * 32 + lane;                 // float4 index in tile
    const int row = f >> 4;
    const int c4  = f & 15;
    const uint32_t ldst = lds_base + (uint32_t)(row * KPITCH + c4 * 4) * 4u;
    const float* ga = gsrc + (size_t)f * 4;
    asm volatile("global_load_async_to_lds_b128 %0, %1, off"
                 :: "v"(ldst), "v"(ga) : "memory");
  }
}

// ---------------- per-batch column sums of V: vtot[b][d] ----------------
__global__ void vtot_kernel(const float* __restrict__ V, float* __restrict__ vtot) {
  const int b = blockIdx.x;
  const int d = threadIdx.x;           // 0..63
  const float* Vb = V + (size_t)b * SEQ * DM;
  float s = 0.f;
#pragma unroll 8
  for (int k = 0; k < SEQ; ++k) s += Vb[(size_t)k * DM + d];
  vtot[b * DM + d] = s;
}

// ---------------- main kernel: one wave handles a 16-query tile ----------------
__global__ void __launch_bounds__(32) banded_attn_kernel(
    const float* __restrict__ Q, const float* __restrict__ K,
    const float* __restrict__ V, const float* __restrict__ vtot,
    float* __restrict__ out)
{
  __shared__ float Klds[2][KBUFSZ];    // double-buffered staged K tiles
  __shared__ float Pt[16 * 16];        // P tile bounce buffer (D-layout -> A-layout)

  const int tile = blockIdx.x;         // 0 .. BATCH*NKT-1
  const int b    = tile >> 8;
  const int tq   = tile & (NKT - 1);
  const int q0   = tq * 16;
  const int lane = threadIdx.x;        // wave32
  const int ln   = lane & 15;          // = column n (B/C/D) and row m (A)
  const int kh   = lane >> 4;

  const uint32_t kbuf0 = (uint32_t)(uintptr_t)&Klds[0][0];
  const uint32_t kbuf1 = (uint32_t)(uintptr_t)&Klds[1][0];

  const float* Qb = Q + ((size_t)b * SEQ + q0) * DM;
  const float* Kb = K + (size_t)b * SEQ * DM;
  const float* Vb = V + (size_t)b * SEQ * DM;

  // ---- Q tile as 16 chained A-fragments over the 64 head dims, scaled 1/8 ----
  v2f qf[16];
#pragma unroll
  for (int kc = 0; kc < 16; ++kc) {
    v2f t = *(const v2f*)(Qb + ln * DM + kc * 4 + kh * 2);
    qf[kc] = t * 0.125f;
  }

  // ================= stage 1: full-row max over ALL 4096 keys =================
  v8f rmax;
#pragma unroll
  for (int r = 0; r < 8; ++r) rmax[r] = -3.4e38f;

  async_stage_tile(Kb, kbuf0, lane);   // prefetch tile 0
  for (int kt = 0; kt < NKT; ++kt) {
    if (kt + 1 < NKT) {
      // prefetch next tile into the other buffer, then wait for current group
      async_stage_tile(Kb + (size_t)(kt + 1) * 16 * DM,
                       (kt & 1) ? kbuf0 : kbuf1, lane);
      asm volatile("s_wait_asynccnt 0x8" ::: "memory");
    } else {
      asm volatile("s_wait_asynccnt 0x0" ::: "memory");
    }
    const float* Kl = &Klds[kt & 1][0];
    v8f d0 = {}, d1 = {};
#pragma unroll
    for (int kc = 0; kc < 16; kc += 2) {
      v2f b0 = *(const v2f*)(Kl + ln * KPITCH + kc * 4 + kh * 2);
      v2f b1 = *(const v2f*)(Kl + ln * KPITCH + (kc + 1) * 4 + kh * 2);
      d0 = wmma4(qf[kc],     b0, d0);
      d1 = wmma4(qf[kc + 1], b1, d1);
    }
#pragma unroll
    for (int r = 0; r < 8; ++r) rmax[r] = fmaxf(rmax[r], d0[r] + d1[r]);
  }
  // cross-lane max over the 16 columns (xor within each 16-lane half)
#pragma unroll
  for (int mask = 1; mask <= 8; mask <<= 1) {
#pragma unroll
    for (int r = 0; r < 8; ++r)
      rmax[r] = fmaxf(rmax[r], __shfl_xor(rmax[r], mask, 32));
  }
  // now rmax[r] == M[row r + 8*kh]  (exactly the rows this lane's D elements use)

  // ================= covered band tiles =================
  const int kt_lo = (tq - 4 < 0) ? 0 : tq - 4;
  const int kt_hi = (tq + 4 > NKT - 1) ? (NKT - 1) : tq + 4;
  const int ncov  = (kt_hi - kt_lo + 1) * 16;

  // ---- pass A: m2 = max over row of a (a = exp(s-M) in band, 0 elsewhere) ----
  v8f am = {};
  for (int kt = kt_lo; kt <= kt_hi; ++kt) {
    async_stage_tile(Kb + (size_t)kt * 16 * DM, kbuf0, lane);
    asm volatile("s_wait_asynccnt 0x0" ::: "memory");
    const float* Kl = &Klds[0][0];
    v8f d0 = {}, d1 = {};
#pragma unroll
    for (int kc = 0; kc < 16; kc += 2) {
      v2f b0 = *(const v2f*)(Kl + ln * KPITCH + kc * 4 + kh * 2);
      v2f b1 = *(const v2f*)(Kl + ln * KPITCH + (kc + 1) * 4 + kh * 2);
      d0 = wmma4(qf[kc],     b0, d0);
      d1 = wmma4(qf[kc + 1], b1, d1);
    }
    const int j = kt * 16 + ln;
#pragma unroll
    for (int r = 0; r < 8; ++r) {
      const int diff = j - (q0 + r + 8 * kh);
      const float sc = d0[r] + d1[r];
      const float a  = (diff >= BAND_LO && diff <= BAND_HI) ? __expf(sc - rmax[r]) : 0.f;
      am[r] = fmaxf(am[r], a);
    }
  }
#pragma unroll
  for (int mask = 1; mask <= 8; mask <<= 1) {
#pragma unroll
    for (int r = 0; r < 8; ++r)
      am[r] = fmaxf(am[r], __shfl_xor(am[r], mask, 32));
  }
  v8f e2;                               // exp(-m2) per row
#pragma unroll
  for (int r = 0; r < 8; ++r) e2[r] = __expf(-am[r]);

  // ---- pass B: P = exp(a - m2) over covered tiles; out_acc = P @ V (WMMA) ----
  v8f acc[4] = {{}, {}, {}, {}};        // out accumulators, d-block nb: d = nb*16 + ln
  v8f zacc = {};                        // sum of p over covered columns
  float vsum[4] = {0.f, 0.f, 0.f, 0.f}; // this lane's half of sum_cov V[k][nb*16+ln]

  for (int kt = kt_lo; kt <= kt_hi; ++kt) {
    const int k0 = kt * 16;
    async_stage_tile(Kb + (size_t)k0 * DM, kbuf0, lane);
    asm volatile("s_wait_asynccnt 0x0" ::: "memory");
    const float* Kl = &Klds[0][0];
    v8f d0 = {}, d1 = {};
#pragma unroll
    for (int kc = 0; kc < 16; kc += 2) {
      v2f b0 = *(const v2f*)(Kl + ln * KPITCH + kc * 4 + kh * 2);
      v2f b1 = *(const v2f*)(Kl + ln * KPITCH + (kc + 1) * 4 + kh * 2);
      d0 = wmma4(qf[kc],     b0, d0);
      d1 = wmma4(qf[kc + 1], b1, d1);
    }
    const int j = k0 + ln;
    v8f p;
#pragma unroll
    for (int r = 0; r < 8; ++r) {
      const int diff = j - (q0 + r + 8 * kh);
      const float sc = d0[r] + d1[r];
      const float a  = (diff >= BAND_LO && diff <= BAND_HI) ? __expf(sc - rmax[r]) : 0.f;
      p[r] = __expf(a - am[r]);         // covered non-band -> exp(-m2), correct
      zacc[r] += p[r];
    }

    // D-layout -> LDS -> A-layout for the P @ V GEMM
    __syncthreads();
#pragma unroll
    for (int r = 0; r < 8; ++r) Pt[(r + 8 * kh) * 16 + ln] = p[r];
    __syncthreads();

    v2f pf[4];
#pragma unroll
    for (int kc = 0; kc < 4; ++kc)
      pf[kc] = *(const v2f*)(&Pt[ln * 16 + kc * 4 + kh * 2]);

#pragma unroll
    for (int nb = 0; nb < 4; ++nb) {
      const float* Vt = Vb + (size_t)k0 * DM + nb * 16;
      v8f dd = acc[nb];
#pragma unroll
      for (int kc = 0; kc < 4; ++kc) {
        const float b0v = Vt[(kc * 4 + kh * 2) * DM + ln];
        const float b1v = Vt[(kc * 4 + kh * 2 + 1) * DM + ln];
        v2f bb; bb[0] = b0v; bb[1] = b1v;
        dd = wmma4(pf[kc], bb, dd);
        vsum[nb] += b0v + b1v;          // partial covered V column sum (half the rows)
      }
      acc[nb] = dd;
    }
  }

  // ---- finalize Z per row: covered sum + (SEQ - ncov) * exp(-m2) ----
#pragma unroll
  for (int mask = 1; mask <= 8; mask <<= 1) {
#pragma unroll
    for (int r = 0; r < 8; ++r)
      zacc[r] += __shfl_xor(zacc[r], mask, 32);
  }
  const float tailn = (float)(SEQ - ncov);
  v8f zrow;
#pragma unroll
  for (int r = 0; r < 8; ++r) zrow[r] = zacc[r] + tailn * e2[r];

  // ---- finalize output: add exp(-m2) * (Vtot - Vcov) tail, divide by Z ----
#pragma unroll
  for (int nb = 0; nb < 4; ++nb) {
    // combine the two half-row partial sums of covered V
    const float vcov = vsum[nb] + __shfl_xor(vsum[nb], 16, 32);
    const float vt   = vtot[b * DM + nb * 16 + ln];
    const float vcorr = vt - vcov;
#pragma unroll
    for (int r = 0; r < 8; ++r) {
      const int row = r + 8 * kh;
      const float o = (acc[nb][r] + e2[r] * vcorr) / zrow[r];
      out[((size_t)b * SEQ + q0 + row) * DM + nb * 16 + ln] = o;
    }
  }
}

extern "C" void kernel_launch(void* const* d_in, const int* in_sizes, int n_in,
                              void* d_out, int out_size, void* d_ws, size_t ws_size,
                              hipStream_t stream) {
  const float* Q = (const float*)d_in[0];
  const float* K = (const float*)d_in[1];
  const float* V = (const float*)d_in[2];
  float* out  = (float*)d_out;
  float* vtot = (float*)d_ws;          // BATCH*DM floats = 1 KB of scratch

  vtot_kernel<<<dim3(BATCH), dim3(DM), 0, stream>>>(V, vtot);
  banded_attn_kernel<<<dim3(BATCH * NKT), dim3(32), 0, stream>>>(Q, K, V, vtot, out);
}